// EdgeOrientation2mer_55576876810304
// MI455X (gfx1250) — compile-verified
//
#include <hip/hip_runtime.h>

// Problem shape (from setup_inputs): B=2, N=2048, K=48, A=4 atoms -> 8 atoms/edge,
// 64 pairs/edge, 192 output floats/edge. Output = 151 MB f32 (bandwidth-bound:
// ~6.5us at 23.3 TB/s). Rotation offloaded to V_WMMA_F32_16X16X4_F32.
#define NB 2
#define NN 2048
#define NK 48
#define NODE_F 12          // floats per node in X (4 atoms * 3)
#define EDGE_F 192         // output floats per edge (64 pairs * 3)

typedef float v2f __attribute__((ext_vector_type(2)));
typedef float v8f __attribute__((ext_vector_type(8)));

__device__ __forceinline__ float3 load3(const float* __restrict__ p) {
    return make_float3(p[0], p[1], p[2]);
}

// reference _normed_vec: v / sqrt(dot(v,v) + 0.1)
__device__ __forceinline__ float3 normed(float3 v) {
    float d = v.x * v.x + v.y * v.y + v.z * v.z + 0.1f;
    float r = __builtin_amdgcn_rsqf(d);          // v_rsq_f32, ~1 ulp
    return make_float3(v.x * r, v.y * r, v.z * r);
}

__device__ __forceinline__ float3 cross3(float3 a, float3 b) {
    return make_float3(a.y * b.z - a.z * b.y,
                       a.z * b.x - a.x * b.z,
                       a.x * b.y - a.y * b.x);
}

// ---------------------------------------------------------------------------
// Stage 1: per-node frame R (rows R[x][y] = n_{y+1}[x]) + mask into workspace.
// Layout per node (12 floats): R00 R01 R02 R10 R11 R12 R20 R21 R22 mask 0 0
// ---------------------------------------------------------------------------
__global__ void frames_kernel(const float* __restrict__ X,
                              const int* __restrict__ C,
                              float* __restrict__ F) {
    int t = blockIdx.x * blockDim.x + threadIdx.x;
    if (t >= NB * NN) return;
    const float* xp = X + (size_t)t * NODE_F;
    float3 xN  = load3(xp + 0);
    float3 xCA = load3(xp + 3);
    float3 xC  = load3(xp + 6);
    float3 n1 = normed(make_float3(xN.x - xCA.x, xN.y - xCA.y, xN.z - xCA.z));
    float3 uc = normed(make_float3(xC.x - xCA.x, xC.y - xCA.y, xC.z - xCA.z));
    float3 n2 = normed(cross3(n1, uc));
    float3 n3 = normed(cross3(n1, n2));
    float* f = F + (size_t)t * 12;
    f[0] = n1.x; f[1] = n2.x; f[2] = n3.x;
    f[3] = n1.y; f[4] = n2.y; f[5] = n3.y;
    f[6] = n1.z; f[7] = n2.z; f[8] = n3.z;
    f[9] = (C[t] > 0) ? 1.0f : 0.0f;
    f[10] = 0.0f; f[11] = 0.0f;
}

// ---------------------------------------------------------------------------
// Stage 2: one wave32 per edge. 64 pairs -> 4x V_WMMA_F32_16X16X4_F32.
//   A (16x4): lane<16 holds (u.x,u.y) of pair m16; lane>=16 holds (u.z, 0).
//   B (4x16): rows of mask*R; row0/row1 in lanes 0-15 (vgpr0/vgpr1),
//             row2/row3(=0) in lanes 16-31. Cols >=3 are zero.
//   D (16x16): cols 0..2 = rotated vectors; repacked via LDS, then two
//              coalesced b128 global stores per edge.
// ---------------------------------------------------------------------------
__launch_bounds__(256)
__global__ void edge_kernel(const float* __restrict__ X,
                            const int* __restrict__ EIDX,
                            const float* __restrict__ F,
                            float* __restrict__ OUT) {
    __shared__ __align__(16) float lds[8 * EDGE_F];   // 6 KB: 8 waves/block

    const int lane = threadIdx.x & 31;
    const int wid  = threadIdx.x >> 5;
    const int e    = blockIdx.x * 8 + wid;            // edge id, grid is exact
    if (e >= NB * NN * NK) return;                    // wave-uniform guard

    const int bn = e / NK;                            // b*NN + n
    const int b  = bn / NN;
    const int j  = EIDX[e];
    const int bj = b * NN + j;

    const float* Xi = X + (size_t)bn * NODE_F;
    const float* Xj = X + (size_t)bj * NODE_F;
    const float* Fi = F + (size_t)bn * 12;
    const float* Fj = F + (size_t)bj * 12;

    const float mask = Fi[9] * Fj[9];                 // mask_i * mask_j

    // ---- B matrix (mask folded in) -------------------------------------
    const int m16 = lane & 15;
    const int hi  = lane >> 4;                        // 0: K=0,1 half; 1: K=2,3 half
    float b0 = 0.0f, b1 = 0.0f;
    if (m16 < 3) {
        if (hi == 0) { b0 = Fi[0 * 3 + m16] * mask;   // row0 = R[0][y]
                       b1 = Fi[1 * 3 + m16] * mask; } // row1 = R[1][y]
        else         { b0 = Fi[2 * 3 + m16] * mask; } // row2 = R[2][y]; row3 = 0
    }
    v2f Bm; Bm[0] = b0; Bm[1] = b1;

    // ---- per-lane atoms: pair p = g*16 + m16; a = p>>3 = 2g+hh; b = p&7 ----
    const int ab = m16 & 7;                           // "b" atom, constant over g
    const int hh = m16 >> 3;
    const float3 vb = load3(((ab < 4) ? Xi : Xj) + (ab & 3) * 3);

    v8f D[4];
    #pragma unroll
    for (int g = 0; g < 4; ++g) {
        const int aa = 2 * g + hh;                    // "a" atom 0..7
        float3 va = load3(((aa < 4) ? Xi : Xj) + (aa & 3) * 3);
        float3 u  = normed(make_float3(vb.x - va.x, vb.y - va.y, vb.z - va.z));
        v2f Am;
        Am[0] = (hi == 0) ? u.x : u.z;
        Am[1] = (hi == 0) ? u.y : 0.0f;
        v8f c = {};
        D[g] = __builtin_amdgcn_wmma_f32_16x16x4_f32(
            /*neg_a=*/false, Am, /*neg_b=*/false, Bm,
            /*c_mod=*/(short)0, c, /*reuse_a=*/false, /*reuse_b=*/false);
    }

    // ---- repack D through LDS: lane (m<8 ? y : 16+y), vgpr r holds pair
    //      (hi*8 + r), component y = m16. Writers: lanes 0-2 and 16-18. ----
    float* w = lds + wid * EDGE_F;
    if (m16 < 3) {
        const int y = m16;
        const int mbase = hi * 8;
        #pragma unroll
        for (int g = 0; g < 4; ++g) {
            #pragma unroll
            for (int r = 0; r < 8; ++r) {
                w[(g * 16 + mbase + r) * 3 + y] = D[g][r];
            }
        }
    }
    // Intra-wave LDS RAW: DS ops are in-order per wave; wait + clobber keeps
    // the compiler from reordering the coalesced reads above the stores.
    asm volatile("s_wait_dscnt 0x0" ::: "memory");

    // ---- coalesced store: 192 floats = 48 float4 chunks -----------------
    const float4* w4 = (const float4*)w;
    float4* o4 = (float4*)(OUT + (size_t)e * EDGE_F);
    o4[lane] = w4[lane];
    if (lane < 16) o4[32 + lane] = w4[32 + lane];
}

// ---------------------------------------------------------------------------
extern "C" void kernel_launch(void* const* d_in, const int* in_sizes, int n_in,
                              void* d_out, int out_size, void* d_ws, size_t ws_size,
                              hipStream_t stream) {
    const float* X    = (const float*)d_in[0];   // (B,N,4,3) f32
    const int*   EIDX = (const int*)d_in[1];     // (B,N,K)  i32
    const int*   C    = (const int*)d_in[2];     // (B,N)    i32
    float* OUT = (float*)d_out;                  // (B,N,K,192) f32
    float* F   = (float*)d_ws;                   // needs NB*NN*12*4 = 192 KB

    frames_kernel<<<(NB * NN + 255) / 256, 256, 0, stream>>>(X, C, F);

    const int E = NB * NN * NK;                  // 196608, divisible by 8
    edge_kernel<<<E / 8, 256, 0, stream>>>(X, EIDX, F, OUT);
}